// LSTMNet_52759378264413
// MI455X (gfx1250) — compile-verified
//
#include <hip/hip_runtime.h>
#include <cstddef>

// Reference dims: T=4096, B=2048, I=2, H=4, L=4; gate order i,f,g,o.
#define T_LEN 4096
#define B_SZ  2048
#define I_SZ  2
#define H_SZ  4
#define L_N   4
#define G_SZ  16   // 4*H gates
#define TILE  16   // batch rows per block == one WMMA tile

typedef float v2f __attribute__((ext_vector_type(2)));
typedef float v8f __attribute__((ext_vector_type(8)));

// Single-wave workgroup: hardware keeps this wave's LDS ops in order (ISA 7.3),
// so cross-lane LDS handoff needs only a *compiler* ordering fence — no
// s_wait_loadcnt/dscnt barrier conservatism, keeps global loads in flight.
#define WAVE_FENCE() asm volatile("" ::: "memory")

// gfx1250 hardware v_tanh_f32.
__device__ __forceinline__ float tanh_f(float v) {
#if __has_builtin(__builtin_amdgcn_tanhf)
    return __builtin_amdgcn_tanhf(v);
#else
    return 2.0f * __builtin_amdgcn_rcpf(1.0f + __expf(-2.0f * v)) - 1.0f;
#endif
}

// One wave handles 16 batch rows through all 4 layers for all 4096 steps.
// gates(16 batch x 16 gate) = A0(16x4) x B0(4x16) + A1(16x4) x B1(4x16) + bias
//   A0 = layer input (x_t padded to 4, or h_{l-1}); A1 = h_l (recurrent)
//   B0/B1 = transposed weights, resident in VGPRs for the whole kernel.
// Activations are applied in the WMMA D layout (gate type is lane-uniform):
//   sigmoid(x) = 0.5*tanh(0.5x)+0.5 ; tanh(g) directly -> per-lane constants.
__global__ __launch_bounds__(32) void lstm4_wmma_f32(
    const float* __restrict__ xg,   // (T,B,I)
    const float* __restrict__ h0g,  // (L,B,H)
    const float* __restrict__ c0g,  // (L,B,H)
    const float* __restrict__ Wih0, // (16,2)
    const float* __restrict__ WihR, // (3,16,4)
    const float* __restrict__ Whh,  // (4,16,4)
    const float* __restrict__ bih,  // (4,16)
    const float* __restrict__ bhh,  // (4,16)
    float* __restrict__ out)        // ys (T,B,H) ++ hT (L,B,H) ++ cT (L,B,H)
{
    __shared__ float hbuf[L_N][TILE][H_SZ]; // per-layer hidden state (persists over t)
    __shared__ float gbuf[TILE][G_SZ];      // activated-gate transpose bounce

    const int lane  = threadIdx.x & 31;
    const int n     = lane & 15;   // WMMA: A row (batch-in-tile) / B,C,D column
    const int khalf = lane >> 4;   // A/B K-half: lanes<16 -> K 0,1 ; lanes>=16 -> K 2,3
    const int kb    = khalf << 1;
    const int tile0 = blockIdx.x * TILE;

    // Per-lane activation constants (gate type = D-layout column = lane n):
    // n in [8,12) is the g-gate -> tanh(x); else sigmoid(x) = 0.5*tanh(0.5x)+0.5
    const bool  isG      = (n >= 8) && (n < 12);
    const float actScale = isG ? 1.0f : 0.5f;
    const float actMul   = isG ? 1.0f : 0.5f;
    const float actAdd   = isG ? 0.0f : 0.5f;

    // ---- Build constant B matrices (weights, K-major over Wcat = [Win ; Whh]) ----
    // Assumed f32 4x16 B layout mirrors the documented 16x4 A layout:
    //   V0: lanes0-15 (K=kb,N=n) ... (see A-matrix table, CDNA5 ISA)
    v2f B0[L_N], B1[L_N];
    v8f biasC[L_N];
    for (int l = 0; l < L_N; ++l) {
        float w0a, w0b;
        if (l == 0) { // x part: K=0..1 valid, K=2..3 zero-padded
            w0a = (kb     < I_SZ) ? Wih0[n * I_SZ + kb]     : 0.0f;
            w0b = (kb + 1 < I_SZ) ? Wih0[n * I_SZ + kb + 1] : 0.0f;
        } else {
            const float* wp = WihR + ((size_t)(l - 1) * G_SZ + n) * H_SZ + kb;
            w0a = wp[0]; w0b = wp[1];
        }
        B0[l].x = w0a; B0[l].y = w0b;
        const float* hp = Whh + ((size_t)l * G_SZ + n) * H_SZ + kb;
        B1[l].x = hp[0]; B1[l].y = hp[1];

        float bs = bih[l * G_SZ + n] + bhh[l * G_SZ + n]; // bias depends only on N -> splat
        v8f bv;
        #pragma unroll
        for (int r = 0; r < 8; ++r) bv[r] = bs;
        biasC[l] = bv;
    }

    // ---- Load initial state: each lane owns (batch eb, hidden pair jb) c-state ----
    const int eb = lane >> 1;        // 0..15
    const int jb = (lane & 1) << 1;  // 0 or 2
    float cst[L_N][2];
    #pragma unroll
    for (int l = 0; l < L_N; ++l) {
        size_t idx = ((size_t)l * B_SZ + tile0 + eb) * H_SZ + jb;
        cst[l][0] = c0g[idx];
        cst[l][1] = c0g[idx + 1];
        hbuf[l][eb][jb]     = h0g[idx];
        hbuf[l][eb][jb + 1] = h0g[idx + 1];
    }
    WAVE_FENCE();

    // Software-pipelined x: x_t lives in registers; the t+1 load is issued at the
    // top of step t and (with no barrier waits) stays in flight across 4 layers.
    const float* xlane = xg + (size_t)(tile0 + n) * I_SZ; // lane's batch row, t=0
    float xr0 = 0.0f, xr1 = 0.0f;
    if (khalf == 0) { xr0 = xlane[0]; xr1 = xlane[1]; }

    for (int t = 0; t < T_LEN; ++t) {
        // Consume this step's x, then immediately start fetching the next one.
        v2f xcur; xcur.x = xr0; xcur.y = xr1;
        if (t + 1 < T_LEN && khalf == 0) {
            const float* xp = xlane + (size_t)(t + 1) * B_SZ * I_SZ;
            xr0 = xp[0]; xr1 = xp[1];
        }
        if (t + 8 < T_LEN) // warm L2 further ahead -> global_prefetch_b8
            __builtin_prefetch(xlane + (size_t)(t + 8) * B_SZ * I_SZ, 0, 1);

        #pragma unroll
        for (int l = 0; l < L_N; ++l) {
            // A0: layer input (K=0..3 of Wcat); A1: recurrent h_l (K=4..7)
            v2f a0;
            if (l == 0) {
                a0 = xcur; // khalf==1 lanes hold stale 0s; B rows 2,3 are zero anyway
            } else {
                a0.x = hbuf[l - 1][n][kb];
                a0.y = hbuf[l - 1][n][kb + 1];
            }
            v2f a1; a1.x = hbuf[l][n][kb]; a1.y = hbuf[l][n][kb + 1];

            v8f acc = biasC[l];
            acc = __builtin_amdgcn_wmma_f32_16x16x4_f32(
                    false, a1, false, B1[l], (short)0, acc, false, false);
            acc = __builtin_amdgcn_wmma_f32_16x16x4_f32(
                    false, a0, false, B0[l], (short)0, acc, false, false);

            // Activate in D layout (8 independent TRANS ops, lane-uniform type),
            // then transpose via LDS. D: VGPR r -> batch (r + 8*khalf), lane -> gate.
            #pragma unroll
            for (int r = 0; r < 8; ++r) {
                float av = tanh_f(actScale * acc[r]);
                gbuf[r + 8 * khalf][n] = __builtin_fmaf(actMul, av, actAdd);
            }
            WAVE_FENCE();

            // Post-transpose cell update: short chain (pk_fma -> tanh -> mul).
            float I0 = gbuf[eb][jb],      I1 = gbuf[eb][jb + 1];
            float F0 = gbuf[eb][4 + jb],  F1 = gbuf[eb][5 + jb];
            float G0 = gbuf[eb][8 + jb],  G1 = gbuf[eb][9 + jb];
            float O0 = gbuf[eb][12 + jb], O1 = gbuf[eb][13 + jb];

            float c0n = __builtin_fmaf(F0, cst[l][0], I0 * G0);
            float c1n = __builtin_fmaf(F1, cst[l][1], I1 * G1);
            cst[l][0] = c0n; cst[l][1] = c1n;
            float h0n = O0 * tanh_f(c0n);
            float h1n = O1 * tanh_f(c1n);

            hbuf[l][eb][jb]     = h0n;
            hbuf[l][eb][jb + 1] = h1n;
            if (l == L_N - 1) { // only the top layer's h ever touches HBM
                size_t yi = ((size_t)t * B_SZ + tile0 + eb) * H_SZ + jb;
                out[yi]     = h0n;
                out[yi + 1] = h1n;
            }
            WAVE_FENCE();
        }
    }

    // ---- Final h/c stacks ----
    float* hs = out + (size_t)T_LEN * B_SZ * H_SZ;
    float* cs = hs  + (size_t)L_N  * B_SZ * H_SZ;
    #pragma unroll
    for (int l = 0; l < L_N; ++l) {
        size_t idx = ((size_t)l * B_SZ + tile0 + eb) * H_SZ + jb;
        hs[idx]     = hbuf[l][eb][jb];
        hs[idx + 1] = hbuf[l][eb][jb + 1];
        cs[idx]     = cst[l][0];
        cs[idx + 1] = cst[l][1];
    }
}

extern "C" void kernel_launch(void* const* d_in, const int* in_sizes, int n_in,
                              void* d_out, int out_size, void* d_ws, size_t ws_size,
                              hipStream_t stream) {
    (void)in_sizes; (void)n_in; (void)out_size; (void)d_ws; (void)ws_size;
    const float* xg   = (const float*)d_in[0];
    const float* h0g  = (const float*)d_in[1];
    const float* c0g  = (const float*)d_in[2];
    const float* Wih0 = (const float*)d_in[3];
    const float* WihR = (const float*)d_in[4];
    const float* Whh  = (const float*)d_in[5];
    const float* bih  = (const float*)d_in[6];
    const float* bhh  = (const float*)d_in[7];
    float* out = (float*)d_out;

    dim3 grid(B_SZ / TILE);  // 128 independent batch tiles
    dim3 block(32);          // one wave32 per tile
    hipLaunchKernelGGL(lstm4_wmma_f32, grid, block, 0, stream,
                       xg, h0g, c0g, Wih0, WihR, Whh, bih, bhh, out);
}